// PatientRepresentationGNN_73366631350607
// MI455X (gfx1250) — compile-verified
//
#include <hip/hip_runtime.h>
#include <hip/hip_bf16.h>

#define DD 128   // input feature dim
#define HD 64    // hidden dim
#define OD 32    // output dim
#define NEG_SLOPE 0.2f

typedef float v2f __attribute__((ext_vector_type(2)));
typedef float v8f __attribute__((ext_vector_type(8)));

// ---------- monotone float<->uint key for atomic float max ----------
__device__ __forceinline__ unsigned f2key(float f) {
  unsigned u = __float_as_uint(f);
  return (u & 0x80000000u) ? ~u : (u | 0x80000000u);
}
__device__ __forceinline__ float key2f(unsigned k) {
  unsigned u = (k & 0x80000000u) ? (k & 0x7FFFFFFFu) : ~k;
  return __uint_as_float(u);
}
__device__ __forceinline__ float leaky(float x) { return x > 0.f ? x : NEG_SLOPE * x; }

// ---------------------------------------------------------------------------
// H = X @ W   (X: [N,128] rm, W: [128,64] rm, H: [N,64] rm), fp32 WMMA.
// One wave -> 16 rows x 64 cols tile; K loop in steps of 4 via
// V_WMMA_F32_16X16X4_F32.  A frag (16x4): lane<16 holds M=lane,K={0,1};
// lane>=16 holds M=lane-16,K={2,3}.  B frag (4x16): lane<16 holds N=lane,
// K={0,1}; lane>=16 holds N=lane-16,K={2,3}.  C/D 16x16: VGPR v, lanes0-15 ->
// (M=v, N=lane); lanes16-31 -> (M=v+8, N=lane-16).
// ---------------------------------------------------------------------------
__global__ void gemm_xw_wmma(const float* __restrict__ X, const float* __restrict__ W,
                             float* __restrict__ Hout, int N) {
  const int lane = threadIdx.x & 31;
  const int wave = threadIdx.x >> 5;
  const int row0 = (blockIdx.x * (int)(blockDim.x >> 5) + wave) * 16;
  if (row0 >= N) return;                    // wave-uniform exit
  const int half = lane >> 4;               // 0 or 1
  const int l16  = lane & 15;
  int arow = row0 + l16; if (arow >= N) arow = N - 1;   // clamp tail (stores guarded)
  const float* __restrict__ xrow = X + (size_t)arow * DD;

  v8f c0 = {}, c1 = {}, c2 = {}, c3 = {};
  for (int k = 0; k < DD; k += 4) {
    const int ka = k + half * 2;
    v2f a; a.x = xrow[ka]; a.y = xrow[ka + 1];
    const float* wr0 = W + (size_t)ka * HD + l16;
    const float* wr1 = wr0 + HD;
    v2f b0; b0.x = wr0[0];  b0.y = wr1[0];
    v2f b1; b1.x = wr0[16]; b1.y = wr1[16];
    v2f b2; b2.x = wr0[32]; b2.y = wr1[32];
    v2f b3; b3.x = wr0[48]; b3.y = wr1[48];
    c0 = __builtin_amdgcn_wmma_f32_16x16x4_f32(false, a, false, b0, (short)0, c0, false, false);
    c1 = __builtin_amdgcn_wmma_f32_16x16x4_f32(false, a, false, b1, (short)0, c1, false, false);
    c2 = __builtin_amdgcn_wmma_f32_16x16x4_f32(false, a, false, b2, (short)0, c2, false, false);
    c3 = __builtin_amdgcn_wmma_f32_16x16x4_f32(false, a, false, b3, (short)0, c3, false, false);
  }
#pragma unroll
  for (int v = 0; v < 8; ++v) {
    const int r = row0 + half * 8 + v;
    if (r < N) {
      float* orow = Hout + (size_t)r * HD + l16;
      orow[0]  = c0[v];
      orow[16] = c1[v];
      orow[32] = c2[v];
      orow[48] = c3[v];
    }
  }
}

// out[row] = dot(H[row,:64], att[:64]); one wave per row
__global__ void rowdot(const float* __restrict__ Hm, const float* __restrict__ att,
                       float* __restrict__ out, int N) {
  const int lane = threadIdx.x & 31;
  const int row = blockIdx.x * (int)(blockDim.x >> 5) + (threadIdx.x >> 5);
  if (row >= N) return;
  const float* hr = Hm + (size_t)row * HD;
  float s = hr[lane] * att[lane] + hr[lane + 32] * att[lane + 32];
#pragma unroll
  for (int off = 16; off > 0; off >>= 1) s += __shfl_xor(s, off, 32);
  if (lane == 0) out[row] = s;
}

// edge pass 1: segment max of leaky_relu(a_s[src]+a_d[dst]) via ordered-key atomicMax
__global__ void edge_alpha_max(const int* __restrict__ src, const int* __restrict__ dst,
                               int E, int m,
                               const float* __restrict__ a_s, const float* __restrict__ a_d,
                               unsigned* __restrict__ amax_key) {
  const int i = blockIdx.x * blockDim.x + threadIdx.x;
  if (i >= E + m) return;
  int s, d; bool valid;
  if (i < E) { s = src[i]; d = dst[i]; valid = (s != d); }   // remove_self_loops
  else       { s = d = i - E; valid = true; }                 // fresh self loops
  if (!valid) return;
  const float al = leaky(a_s[s] + a_d[d]);
  atomicMax(amax_key + d, f2key(al));
}

// edge pass 2: denom[d] += exp(alpha - amax[d])
__global__ void edge_denom(const int* __restrict__ src, const int* __restrict__ dst,
                           int E, int m,
                           const float* __restrict__ a_s, const float* __restrict__ a_d,
                           const unsigned* __restrict__ amax_key, float* __restrict__ denom) {
  const int i = blockIdx.x * blockDim.x + threadIdx.x;
  if (i >= E + m) return;
  int s, d; bool valid;
  if (i < E) { s = src[i]; d = dst[i]; valid = (s != d); }
  else       { s = d = i - E; valid = true; }
  if (!valid) return;
  const float al = leaky(a_s[s] + a_d[d]);
  atomicAdd(denom + d, expf(al - key2f(amax_key[d])));
}

// edge pass 3: acc[d,:] += h_src[s,:] * (exp(alpha-amax)/max(denom,1e-16)); 64 lanes/edge
__global__ void edge_aggregate(const int* __restrict__ src, const int* __restrict__ dst,
                               int E, int m,
                               const float* __restrict__ a_s, const float* __restrict__ a_d,
                               const unsigned* __restrict__ amax_key,
                               const float* __restrict__ denom,
                               const float* __restrict__ Hsrc, float* __restrict__ acc) {
  const int t = blockIdx.x * blockDim.x + threadIdx.x;
  const int e = t >> 6;
  const int f = t & 63;
  if (e >= E + m) return;
  int s, d; bool valid;
  if (e < E) { s = src[e]; d = dst[e]; valid = (s != d); }
  else       { s = d = e - E; valid = true; }
  if (!valid) return;
  const float al = leaky(a_s[s] + a_d[d]);
  const float ex = expf(al - key2f(amax_key[d]));
  float dn = denom[d]; dn = dn > 1e-16f ? dn : 1e-16f;
  const float w = ex / dn;
  atomicAdd(acc + (size_t)d * HD + f, Hsrc[(size_t)s * HD + f] * w);
}

// out[p,:32] = elu(acc[p,:64] + b0+b1+b2) @ lin_w + lin_b ; one wave per patient
__global__ void final_linear(const float* __restrict__ acc,
                             const float* __restrict__ b0, const float* __restrict__ b1,
                             const float* __restrict__ b2,
                             const float* __restrict__ lin_w, const float* __restrict__ lin_b,
                             float* __restrict__ out, int NP) {
  const int lane = threadIdx.x & 31;
  const int p = blockIdx.x * (int)(blockDim.x >> 5) + (threadIdx.x >> 5);
  if (p >= NP) return;
  float z0 = acc[(size_t)p * HD + lane]      + b0[lane]      + b1[lane]      + b2[lane];
  float z1 = acc[(size_t)p * HD + lane + 32] + b0[lane + 32] + b1[lane + 32] + b2[lane + 32];
  z0 = z0 > 0.f ? z0 : (expf(z0) - 1.f);     // ELU (alpha=1)
  z1 = z1 > 0.f ? z1 : (expf(z1) - 1.f);
  float s = lin_b[lane];
#pragma unroll
  for (int h = 0; h < 32; ++h) s += __shfl(z0, h, 32) * lin_w[h * OD + lane];
#pragma unroll
  for (int h = 0; h < 32; ++h) s += __shfl(z1, h, 32) * lin_w[(h + 32) * OD + lane];
  out[(size_t)p * OD + lane] = s;
}

// ---------------------------------------------------------------------------
extern "C" void kernel_launch(void* const* d_in, const int* in_sizes, int n_in,
                              void* d_out, int out_size, void* d_ws, size_t ws_size,
                              hipStream_t stream) {
  (void)n_in; (void)out_size; (void)ws_size;
  const float* x_r   = (const float*)d_in[0];
  const float* x_g   = (const float*)d_in[1];
  const float* x_p   = (const float*)d_in[2];
  const int* src_rp  = (const int*)d_in[3];
  const int* dst_rp  = (const int*)d_in[4];
  const int* src_gp  = (const int*)d_in[5];
  const int* dst_gp  = (const int*)d_in[6];
  const int* src_pp  = (const int*)d_in[7];
  const int* dst_pp  = (const int*)d_in[8];
  const float* W_rp  = (const float*)d_in[9];
  const float* att_s_rp = (const float*)d_in[10];
  const float* att_d_rp = (const float*)d_in[11];
  const float* b_rp  = (const float*)d_in[12];
  const float* W_gp  = (const float*)d_in[13];
  const float* att_s_gp = (const float*)d_in[14];
  const float* att_d_gp = (const float*)d_in[15];
  const float* b_gp  = (const float*)d_in[16];
  const float* W_pp  = (const float*)d_in[17];
  const float* att_s_pp = (const float*)d_in[18];
  const float* att_d_pp = (const float*)d_in[19];
  const float* b_pp  = (const float*)d_in[20];
  const float* lin_w = (const float*)d_in[21];
  const float* lin_b = (const float*)d_in[22];
  float* out = (float*)d_out;

  const int NR = in_sizes[0] / DD;
  const int NG = in_sizes[1] / DD;
  const int NP = in_sizes[2] / DD;
  const int ERP = in_sizes[3];
  const int EGP = in_sizes[5];
  const int EPP = in_sizes[7];
  const int m_rp = (NR < NP) ? NR : NP;
  const int m_gp = (NG < NP) ? NG : NP;
  const int m_pp = NP;

  // ---- carve workspace (~60 MB) ----
  char* ws = (char*)d_ws;
  size_t off = 0;
  auto carve = [&](size_t bytes) -> char* {
    char* p = ws + off;
    off = (off + bytes + 255) & ~(size_t)255;
    return p;
  };
  float*    h_r    = (float*)carve((size_t)NR * HD * 4);
  float*    h_g    = (float*)carve((size_t)NG * HD * 4);
  float*    h_p_rp = (float*)carve((size_t)NP * HD * 4);
  float*    h_p_gp = (float*)carve((size_t)NP * HD * 4);
  float*    h_p_pp = (float*)carve((size_t)NP * HD * 4);
  float*    acc    = (float*)carve((size_t)NP * HD * 4);
  float*    as_rp  = (float*)carve((size_t)NR * 4);
  float*    as_gp  = (float*)carve((size_t)NG * 4);
  float*    as_pp  = (float*)carve((size_t)NP * 4);
  float*    ad_rp  = (float*)carve((size_t)NP * 4);
  float*    ad_gp  = (float*)carve((size_t)NP * 4);
  float*    ad_pp  = (float*)carve((size_t)NP * 4);
  unsigned* mx_rp  = (unsigned*)carve((size_t)NP * 4);
  unsigned* mx_gp  = (unsigned*)carve((size_t)NP * 4);
  unsigned* mx_pp  = (unsigned*)carve((size_t)NP * 4);
  float*    dn_rp  = (float*)carve((size_t)NP * 4);
  float*    dn_gp  = (float*)carve((size_t)NP * 4);
  float*    dn_pp  = (float*)carve((size_t)NP * 4);

  // ---- zero-init reduction buffers (every call; graph-capture safe) ----
  hipMemsetAsync(acc,   0, (size_t)NP * HD * 4, stream);
  hipMemsetAsync(mx_rp, 0, (size_t)NP * 4, stream);
  hipMemsetAsync(mx_gp, 0, (size_t)NP * 4, stream);
  hipMemsetAsync(mx_pp, 0, (size_t)NP * 4, stream);
  hipMemsetAsync(dn_rp, 0, (size_t)NP * 4, stream);
  hipMemsetAsync(dn_gp, 0, (size_t)NP * 4, stream);
  hipMemsetAsync(dn_pp, 0, (size_t)NP * 4, stream);

  const dim3 blk(256);
  auto ggrid = [](int n) { return dim3((unsigned)((n + 127) / 128)); };   // 8 waves x 16 rows
  auto wgrid = [](int n) { return dim3((unsigned)((n + 7) / 8)); };       // 8 waves / block
  auto egrid = [](int tot) { return dim3((unsigned)((tot + 255) / 256)); };
  auto agrid = [](int tot) { return dim3((unsigned)(((long long)tot * 64 + 255) / 256)); };

  // ---- projections (WMMA fp32) ----
  gemm_xw_wmma<<<ggrid(NR), blk, 0, stream>>>(x_r, W_rp, h_r,    NR);
  gemm_xw_wmma<<<ggrid(NG), blk, 0, stream>>>(x_g, W_gp, h_g,    NG);
  gemm_xw_wmma<<<ggrid(NP), blk, 0, stream>>>(x_p, W_rp, h_p_rp, NP);
  gemm_xw_wmma<<<ggrid(NP), blk, 0, stream>>>(x_p, W_gp, h_p_gp, NP);
  gemm_xw_wmma<<<ggrid(NP), blk, 0, stream>>>(x_p, W_pp, h_p_pp, NP);

  // ---- attention scalars ----
  rowdot<<<wgrid(NR), blk, 0, stream>>>(h_r,    att_s_rp, as_rp, NR);
  rowdot<<<wgrid(NG), blk, 0, stream>>>(h_g,    att_s_gp, as_gp, NG);
  rowdot<<<wgrid(NP), blk, 0, stream>>>(h_p_pp, att_s_pp, as_pp, NP);
  rowdot<<<wgrid(NP), blk, 0, stream>>>(h_p_rp, att_d_rp, ad_rp, NP);
  rowdot<<<wgrid(NP), blk, 0, stream>>>(h_p_gp, att_d_gp, ad_gp, NP);
  rowdot<<<wgrid(NP), blk, 0, stream>>>(h_p_pp, att_d_pp, ad_pp, NP);

  // ---- relation rp ----
  edge_alpha_max<<<egrid(ERP + m_rp), blk, 0, stream>>>(src_rp, dst_rp, ERP, m_rp, as_rp, ad_rp, mx_rp);
  edge_denom    <<<egrid(ERP + m_rp), blk, 0, stream>>>(src_rp, dst_rp, ERP, m_rp, as_rp, ad_rp, mx_rp, dn_rp);
  edge_aggregate<<<agrid(ERP + m_rp), blk, 0, stream>>>(src_rp, dst_rp, ERP, m_rp, as_rp, ad_rp, mx_rp, dn_rp, h_r, acc);
  // ---- relation gp ----
  edge_alpha_max<<<egrid(EGP + m_gp), blk, 0, stream>>>(src_gp, dst_gp, EGP, m_gp, as_gp, ad_gp, mx_gp);
  edge_denom    <<<egrid(EGP + m_gp), blk, 0, stream>>>(src_gp, dst_gp, EGP, m_gp, as_gp, ad_gp, mx_gp, dn_gp);
  edge_aggregate<<<agrid(EGP + m_gp), blk, 0, stream>>>(src_gp, dst_gp, EGP, m_gp, as_gp, ad_gp, mx_gp, dn_gp, h_g, acc);
  // ---- relation pp ----
  edge_alpha_max<<<egrid(EPP + m_pp), blk, 0, stream>>>(src_pp, dst_pp, EPP, m_pp, as_pp, ad_pp, mx_pp);
  edge_denom    <<<egrid(EPP + m_pp), blk, 0, stream>>>(src_pp, dst_pp, EPP, m_pp, as_pp, ad_pp, mx_pp, dn_pp);
  edge_aggregate<<<agrid(EPP + m_pp), blk, 0, stream>>>(src_pp, dst_pp, EPP, m_pp, as_pp, ad_pp, mx_pp, dn_pp, h_p_pp, acc);

  // ---- ELU + final linear ----
  final_linear<<<wgrid(NP), blk, 0, stream>>>(acc, b_rp, b_gp, b_pp, lin_w, lin_b, out, NP);
}